// Three_D_Branch_26508538151513
// MI455X (gfx1250) — compile-verified
//
#include <hip/hip_runtime.h>
#include <hip/hip_bf16.h>
#include <float.h>

#define B_ 2
#define N_ 8192
#define C_ 256
#define H_ 256
#define W_ 512
#define K_ 3

typedef float v2f __attribute__((ext_vector_type(2)));
typedef float v8f __attribute__((ext_vector_type(8)));

// ---------------------------------------------------------------------------
// Kernel 1: projection + channel gather + lidar passthrough.
// One block (256 threads) per point; thread c handles channel c.
// ---------------------------------------------------------------------------
__global__ __launch_bounds__(256) void gather_kernel(
    const float* __restrict__ features,   // (B, C, H, W)
    const float* __restrict__ lidar,      // (B, N, 3)
    const float* __restrict__ P,          // (3, 4)
    float* __restrict__ outF,             // (B, N, C)
    float* __restrict__ outL)             // (B, N, 3)
{
    const int bn = blockIdx.x;            // 0 .. B*N-1
    const int b  = bn / N_;
    const int c  = threadIdx.x;

    const float* p = lidar + (size_t)bn * 3;
    const float x = p[0], y = p[1], z = p[2];

    // cam = P * (x, y, z, 1)
    const float cam0 = P[0]*x + P[1]*y + P[2]*z  + P[3];
    const float cam1 = P[4]*x + P[5]*y + P[6]*z  + P[7];
    const float cam2 = P[8]*x + P[9]*y + P[10]*z + P[11];

    const float px = cam0 / cam2;
    const float py = cam1 / cam2;

    int xi = (int)rintf(px); xi = min(max(xi, 0), W_ - 1);
    int yi = (int)rintf(py); yi = min(max(yi, 0), H_ - 1);

    const size_t src = (((size_t)b * C_ + c) * H_ + yi) * W_ + xi;
    outF[(size_t)bn * C_ + c] = features[src];

    if (threadIdx.x < 3)
        outL[(size_t)bn * 3 + threadIdx.x] = p[threadIdx.x];
}

// ---------------------------------------------------------------------------
// Kernel 2: KNN via V_WMMA_F32_16X16X4_F32 distance tiles + top-4 selection.
// ---------------------------------------------------------------------------
#define CHUNK 4096            // points staged in LDS per pass (64 KB of float4)
#define WAVES_PER_BLOCK 8     // 256 threads, wave32

// Branchless sorted insert, strict < (ties keep earlier=lower index because the
// per-lane candidate stream is ascending in index). Pure v_cndmask codegen.
__device__ __forceinline__ void topk_insert_fast(float (&d)[4], int (&ix)[4],
                                                 float nd, int ni)
{
    const bool b0 = nd < d[0];
    const bool b1 = nd < d[1];
    const bool b2 = nd < d[2];
    const bool b3 = nd < d[3];
    const float e0 = b0 ? nd : d[0];
    const int   j0 = b0 ? ni : ix[0];
    const float e1 = b1 ? (b0 ? d[0] : nd) : d[1];
    const int   j1 = b1 ? (b0 ? ix[0] : ni) : ix[1];
    const float e2 = b2 ? (b1 ? d[1] : nd) : d[2];
    const int   j2 = b2 ? (b1 ? ix[1] : ni) : ix[2];
    const float e3 = b3 ? (b2 ? d[2] : nd) : d[3];
    const int   j3 = b3 ? (b2 ? ix[2] : ni) : ix[3];
    d[0] = e0; d[1] = e1; d[2] = e2; d[3] = e3;
    ix[0] = j0; ix[1] = j1; ix[2] = j2; ix[3] = j3;
}

// Branchless sorted insert with lowest-index tie-break (cross-lane merge).
__device__ __forceinline__ void topk_insert_tb(float (&d)[4], int (&ix)[4],
                                               float nd, int ni)
{
    const bool b0 = (nd < d[0]) || (nd == d[0] && ni < ix[0]);
    const bool b1 = (nd < d[1]) || (nd == d[1] && ni < ix[1]);
    const bool b2 = (nd < d[2]) || (nd == d[2] && ni < ix[2]);
    const bool b3 = (nd < d[3]) || (nd == d[3] && ni < ix[3]);
    const float e0 = b0 ? nd : d[0];
    const int   j0 = b0 ? ni : ix[0];
    const float e1 = b1 ? (b0 ? d[0] : nd) : d[1];
    const int   j1 = b1 ? (b0 ? ix[0] : ni) : ix[1];
    const float e2 = b2 ? (b1 ? d[1] : nd) : d[2];
    const int   j2 = b2 ? (b1 ? ix[1] : ni) : ix[2];
    const float e3 = b3 ? (b2 ? d[2] : nd) : d[3];
    const int   j3 = b3 ? (b2 ? ix[2] : ni) : ix[3];
    d[0] = e0; d[1] = e1; d[2] = e2; d[3] = e3;
    ix[0] = j0; ix[1] = j1; ix[2] = j2; ix[3] = j3;
}

__global__ __launch_bounds__(256) void knn_kernel(
    const float* __restrict__ lidar,   // (B, N, 3)
    float* __restrict__ outK)          // (B, N, K) as float-valued indices
{
    __shared__ float4 pts[CHUNK];      // (-2x, -2y, -2z, |p|^2)

    const int blocksPerBatch = N_ / (16 * WAVES_PER_BLOCK); // 64
    const int b    = blockIdx.x / blocksPerBatch;
    const int qg   = blockIdx.x % blocksPerBatch;
    const int wave = threadIdx.x >> 5;
    const int lane = threadIdx.x & 31;
    const int lm   = lane & 15;
    const int hi   = lane >> 4;                // 0: K0/K1 half, 1: K2/K3 half
    const int qbase = (qg * WAVES_PER_BLOCK + wave) * 16;

    // ---- A fragment (loop-invariant): query row = (qx, qy, qz, 1) ----
    const float* qp = lidar + ((size_t)b * N_ + qbase + lm) * 3;
    const float qx = qp[0], qy = qp[1], qz = qp[2];
    v2f a;
    a.x = hi ? qz   : qx;
    a.y = hi ? 1.0f : qy;

    // ---- C operand (loop-invariant): C[m][n] = |q_m|^2 ----
    // Lane holds rows M = r + 8*hi, so it needs |q|^2 of 8 queries.
    v8f cc;
    #pragma unroll
    for (int r = 0; r < 8; ++r) {
        const float* qq = lidar + ((size_t)b * N_ + qbase + r + 8 * hi) * 3;
        const float X = qq[0], Y = qq[1], Z = qq[2];
        cc[r] = X * X + Y * Y + Z * Z;
    }

    // Per-lane running top-4 per owned row (rows M = r + 8*hi, column = lm-residue)
    float bd[8][4];
    int   bi[8][4];
    #pragma unroll
    for (int r = 0; r < 8; ++r)
        #pragma unroll
        for (int j = 0; j < 4; ++j) { bd[r][j] = FLT_MAX; bi[r][j] = 0x7fffffff; }

    // Conservative per-lane rejection threshold: worst >= max_r bd[r][3].
    float worst = FLT_MAX;

    for (int base = 0; base < N_; base += CHUNK) {
        __syncthreads();
        for (int i = threadIdx.x; i < CHUNK; i += 256) {
            const float* pp = lidar + ((size_t)b * N_ + base + i) * 3;
            const float x = pp[0], y = pp[1], z = pp[2];
            pts[i] = make_float4(-2.0f * x, -2.0f * y, -2.0f * z,
                                 x * x + y * y + z * z);
        }
        __syncthreads();

        const int TILES = CHUNK / 16;  // 256
        // Software pipeline: current pair in regs, prefetch 2 ahead.
        float4 c0 = pts[lm];
        float4 c1 = pts[16 + lm];
        for (int t = 0; t < TILES; t += 2) {
            const int tn = (t + 2 < TILES) ? (t + 2) : (TILES - 2);
            const float4 n0 = pts[tn * 16 + lm];          // ds_load_b128 (early)
            const float4 n1 = pts[(tn + 1) * 16 + lm];    // ds_load_b128 (early)

            // ---- B fragments: point col = (-2x, -2y, -2z, |p|^2) ----
            v2f f0, f1;
            f0.x = hi ? c0.z : c0.x;  f0.y = hi ? c0.w : c0.y;
            f1.x = hi ? c1.z : c1.x;  f1.y = hi ? c1.w : c1.y;

            // D = |q|^2 + |p|^2 - 2 q.p  (full distance tile in one WMMA)
            v8f dm0 = __builtin_amdgcn_wmma_f32_16x16x4_f32(
                false, a, false, f0, (short)0, cc, false, false);
            v8f dm1 = __builtin_amdgcn_wmma_f32_16x16x4_f32(
                false, a, false, f1, (short)0, cc, false, false);

            // Pure-VALU conservative filter: min over all 16 candidates
            // vs cached max-of-thresholds. False positives are harmless.
            float cmin = fminf(dm0[0], dm1[0]);
            #pragma unroll
            for (int r = 1; r < 8; ++r)
                cmin = fminf(cmin, fminf(dm0[r], dm1[r]));

            if (cmin < worst) {
                const int col0 = base + t * 16 + lm;
                const int col1 = col0 + 16;
                #pragma unroll
                for (int r = 0; r < 8; ++r) {
                    topk_insert_fast(bd[r], bi[r], dm0[r], col0);
                    topk_insert_fast(bd[r], bi[r], dm1[r], col1);
                }
                worst = bd[0][3];
                #pragma unroll
                for (int r = 1; r < 8; ++r) worst = fmaxf(worst, bd[r][3]);
            }
            c0 = n0;
            c1 = n1;
        }
    }

    // ---- butterfly merge of top-4 lists across the 16 lanes sharing each row ----
    #pragma unroll
    for (int off = 1; off < 16; off <<= 1) {
        #pragma unroll
        for (int r = 0; r < 8; ++r) {
            float pd0 = __shfl_xor(bd[r][0], off, 32);
            float pd1 = __shfl_xor(bd[r][1], off, 32);
            float pd2 = __shfl_xor(bd[r][2], off, 32);
            float pd3 = __shfl_xor(bd[r][3], off, 32);
            int   pi0 = __shfl_xor(bi[r][0], off, 32);
            int   pi1 = __shfl_xor(bi[r][1], off, 32);
            int   pi2 = __shfl_xor(bi[r][2], off, 32);
            int   pi3 = __shfl_xor(bi[r][3], off, 32);
            topk_insert_tb(bd[r], bi[r], pd0, pi0);
            topk_insert_tb(bd[r], bi[r], pd1, pi1);
            topk_insert_tb(bd[r], bi[r], pd2, pi2);
            topk_insert_tb(bd[r], bi[r], pd3, pi3);
        }
    }

    // ---- write neighbors 1..3 (0 is the self match, matching idx[:, :, 1:]) ----
    if (lm == 0) {
        #pragma unroll
        for (int r = 0; r < 8; ++r) {
            const int qn = qbase + r + 8 * hi;
            const size_t o = ((size_t)b * N_ + qn) * K_;
            outK[o + 0] = (float)bi[r][1];
            outK[o + 1] = (float)bi[r][2];
            outK[o + 2] = (float)bi[r][3];
        }
    }
}

// ---------------------------------------------------------------------------
extern "C" void kernel_launch(void* const* d_in, const int* in_sizes, int n_in,
                              void* d_out, int out_size, void* d_ws, size_t ws_size,
                              hipStream_t stream)
{
    const float* features = (const float*)d_in[0];
    const float* lidar    = (const float*)d_in[1];
    const float* P        = (const float*)d_in[2];
    // d_in[3] = imgs (unused by the reference outputs)

    float* out  = (float*)d_out;
    float* outF = out;                                  // B*N*C
    float* outL = outF + (size_t)B_ * N_ * C_;          // B*N*3
    float* outK = outL + (size_t)B_ * N_ * 3;           // B*N*K

    gather_kernel<<<B_ * N_, 256, 0, stream>>>(features, lidar, P, outF, outL);

    const int knn_blocks = B_ * (N_ / (16 * WAVES_PER_BLOCK)); // 128
    knn_kernel<<<knn_blocks, 256, 0, stream>>>(lidar, outK);
}